// LogFullNeuralBackflow_9285719294232
// MI455X (gfx1250) — compile-verified
//
#include <hip/hip_runtime.h>
#include <hip/hip_bf16.h>
#include <math.h>

// Problem constants (from reference)
#define BB   16384   // Monte-Carlo samples
#define MM   128     // spin orbitals
#define NF   32      // fermions
#define HH   256     // hidden units
#define CC   (MM*NF) // 4096 output cols of GEMM2

typedef __attribute__((ext_vector_type(2))) float v2f;
typedef __attribute__((ext_vector_type(8))) float v8f;

// ---------------------------------------------------------------------------
// Kernel 1: rank[b][m] = # occupied orbitals with index < m  (stable gather
// order == argsort(-n) of the reference). One thread per sample.
// ---------------------------------------------------------------------------
__global__ __launch_bounds__(256) void k_rank(const float* __restrict__ n,
                                              unsigned char* __restrict__ rank) {
  int b = blockIdx.x * blockDim.x + threadIdx.x;
  if (b >= BB) return;
  const float* nb = n + (size_t)b * MM;
  unsigned char* rb = rank + (size_t)b * MM;
  int cnt = 0;
  for (int m = 0; m < MM; ++m) {
    rb[m] = (unsigned char)cnt;
    if (nb[m] > 0.5f) ++cnt;
  }
}

// ---------------------------------------------------------------------------
// Kernel 2: h = tanh(n @ W1 + b1), f32 WMMA 16x16x4.
// One wave per 16(samples)x16(hidden) tile, K = 128 -> 32 WMMAs.
// grid = (HH/16/8, BB/16), block = 256 (8 waves).
// ---------------------------------------------------------------------------
__global__ __launch_bounds__(256) void k_gemm1(const float* __restrict__ n,
                                               const float* __restrict__ W1,
                                               const float* __restrict__ b1,
                                               float* __restrict__ h) {
  const int lane = threadIdx.x & 31;
  const int wave = threadIdx.x >> 5;
  const int h0 = (blockIdx.x * 8 + wave) * 16;   // hidden-col tile
  const int b0 = blockIdx.y * 16;                // sample-row tile
  const int rowA = lane & 15;                    // M index this lane supplies
  const int koff = (lane >> 4) * 2;              // K sub-pair per ISA layout
  const int colB = lane & 15;                    // N index this lane supplies

  v8f acc = {};
  const float* nrow = n + (size_t)(b0 + rowA) * MM + koff;
  for (int kk = 0; kk < MM; kk += 4) {
    float2 av = *(const float2*)(nrow + kk);          // A: M=rowA, K=kk+koff(+1)
    v2f a; a.x = av.x; a.y = av.y;
    const int kb = kk + koff;
    v2f bm;                                           // B: K=kb(+1), N=colB
    bm.x = W1[(size_t)kb * HH + h0 + colB];
    bm.y = W1[(size_t)(kb + 1) * HH + h0 + colB];
    acc = __builtin_amdgcn_wmma_f32_16x16x4_f32(false, a, false, bm,
                                                (short)0, acc, false, false);
  }
  const float bias = b1[h0 + colB];
  const int rbase = b0 + (lane >> 4) * 8;            // C/D: VGPR v -> M = v + 8*(lane>>4)
#pragma unroll
  for (int v = 0; v < 8; ++v) {
    h[(size_t)(rbase + v) * HH + h0 + colB] = tanhf(acc[v] + bias);
  }
}

// ---------------------------------------------------------------------------
// Kernel 3: F = h @ W2 + b2 computed tile-wise; epilogue scatters only the
// occupied rows straight into A[b, rank[b][m], j]  (full F never hits HBM).
// One wave per 16(samples) x 64(cols) macro-tile: 4 accumulators share one
// A-operand load -> 4x fewer h-panel loads, wmma:vmem issue ratio ~4:9
// instead of 1:3.  K = 256 -> 256 WMMAs per wave.
// grid = (CC/64/8, BB/16), block = 256 (8 waves).
// ---------------------------------------------------------------------------
__global__ __launch_bounds__(256) void k_gemm2(const float* __restrict__ h,
                                               const float* __restrict__ W2,
                                               const float* __restrict__ b2,
                                               const float* __restrict__ n,
                                               const unsigned char* __restrict__ rank,
                                               float* __restrict__ A) {
  const int lane = threadIdx.x & 31;
  const int wave = threadIdx.x >> 5;
  const int c0 = (blockIdx.x * 8 + wave) * 64;   // F-column macro-tile (64 wide)
  const int b0 = blockIdx.y * 16;                // sample-row tile
  const int rowA = lane & 15;
  const int koff = (lane >> 4) * 2;
  const int colB = lane & 15;

  v8f acc0 = {}, acc1 = {}, acc2 = {}, acc3 = {};
  const float* hrow = h + (size_t)(b0 + rowA) * HH + koff;
  for (int kk = 0; kk < HH; kk += 4) {
    float2 av = *(const float2*)(hrow + kk);         // shared A operand
    v2f a; a.x = av.x; a.y = av.y;
    const int kb = kk + koff;
    const float* w0 = W2 + (size_t)kb * CC + c0 + colB;
    const float* w1 = W2 + (size_t)(kb + 1) * CC + c0 + colB;
    v2f bm;
    bm.x = w0[0];  bm.y = w1[0];
    acc0 = __builtin_amdgcn_wmma_f32_16x16x4_f32(false, a, false, bm,
                                                 (short)0, acc0, false, false);
    bm.x = w0[16]; bm.y = w1[16];
    acc1 = __builtin_amdgcn_wmma_f32_16x16x4_f32(false, a, false, bm,
                                                 (short)0, acc1, false, false);
    bm.x = w0[32]; bm.y = w1[32];
    acc2 = __builtin_amdgcn_wmma_f32_16x16x4_f32(false, a, false, bm,
                                                 (short)0, acc2, false, false);
    bm.x = w0[48]; bm.y = w1[48];
    acc3 = __builtin_amdgcn_wmma_f32_16x16x4_f32(false, a, false, bm,
                                                 (short)0, acc3, false, false);
  }

  const int rbase = b0 + (lane >> 4) * 8;
  v8f accs[4] = {acc0, acc1, acc2, acc3};
#pragma unroll
  for (int t = 0; t < 4; ++t) {
    const int c = c0 + t * 16 + colB;
    const float bias = b2[c];
    const int m = c >> 5;        // orbital index
    const int j = c & 31;        // column within 32x32 determinant matrix
#pragma unroll
    for (int v = 0; v < 8; ++v) {
      const int b = rbase + v;
      if (n[(size_t)b * MM + m] > 0.5f) {            // orbital occupied -> keep
        const int i = (int)rank[(size_t)b * MM + m]; // row of A (ascending order)
        A[(size_t)b * (NF * NF) + i * NF + j] = accs[t][v] + bias;
      }
    }
  }
}

// ---------------------------------------------------------------------------
// Kernel 4: slogdet of each 32x32 via partial-pivot LU. One wave per matrix,
// lane r owns row r; matrix lives in LDS with stride-33 padding (64 banks,
// stride 33 words -> conflict-free column access; same-address pivot-row
// reads broadcast for free). 8 waves/block, uniform trip counts so
// __syncthreads() is safe and orders the cross-lane LDS traffic.
// Output: complex64 interleaved (log|det|, 0 or pi).
// ---------------------------------------------------------------------------
__global__ __launch_bounds__(256) void k_slogdet(const float* __restrict__ A,
                                                 float* __restrict__ out) {
  __shared__ float smem[8 * 32 * 33];
  const int lane = threadIdx.x & 31;
  const int wave = threadIdx.x >> 5;
  const int b = blockIdx.x * 8 + wave;
  float* sm = smem + wave * (32 * 33);

  const float* Ab = A + (size_t)b * (NF * NF) + (size_t)lane * NF;
  for (int j = 0; j < 32; ++j) sm[lane * 33 + j] = Ab[j];
  __syncthreads();

  float logabs = 0.0f;
  int neg = 0;
  for (int k = 0; k < 32; ++k) {
    // --- wave argmax over |column k| among rows >= k ---
    float val = (lane >= k) ? fabsf(sm[lane * 33 + k]) : -1.0f;
    int idx = lane;
    for (int off = 16; off > 0; off >>= 1) {
      float ov = __shfl_xor(val, off, 32);
      int oi = __shfl_xor(idx, off, 32);
      if (ov > val || (ov == val && oi < idx)) { val = ov; idx = oi; }
    }
    // --- row swap (lanes act as column index), uniform branch ---
    if (idx != k) {
      float t1 = sm[k * 33 + lane];
      float t2 = sm[idx * 33 + lane];
      sm[k * 33 + lane] = t2;
      sm[idx * 33 + lane] = t1;
      neg ^= 1;
    }
    __syncthreads();
    const float piv = sm[k * 33 + k];
    if (piv < 0.0f) neg ^= 1;
    logabs += logf(fabsf(piv));
    // --- rank-1 elimination update, lane r updates its own row ---
    if (lane > k) {
      const float mlt = sm[lane * 33 + k] / piv;
      for (int j = k + 1; j < 32; ++j)
        sm[lane * 33 + j] = fmaf(-mlt, sm[k * 33 + j], sm[lane * 33 + j]);
    }
    __syncthreads();
  }
  if (lane == 0) {
    out[2 * b + 0] = logabs;                                  // log|det|
    out[2 * b + 1] = neg ? 3.14159265358979323846f : 0.0f;    // arg(sign)
  }
}

// ---------------------------------------------------------------------------
// Launch
// ---------------------------------------------------------------------------
extern "C" void kernel_launch(void* const* d_in, const int* in_sizes, int n_in,
                              void* d_out, int out_size, void* d_ws, size_t ws_size,
                              hipStream_t stream) {
  const float* n  = (const float*)d_in[0];   // [B, M]
  const float* W1 = (const float*)d_in[1];   // [M, H]
  const float* b1 = (const float*)d_in[2];   // [H]
  const float* W2 = (const float*)d_in[3];   // [H, M*NF]
  const float* b2 = (const float*)d_in[4];   // [M*NF]
  float* out = (float*)d_out;                // complex64[B] as interleaved floats

  // workspace: h (16 MB) | rank u8 (2 MB) | A (64 MB)
  char* ws = (char*)d_ws;
  float*         h_buf  = (float*)(ws);
  unsigned char* rank   = (unsigned char*)(ws + (size_t)BB * HH * sizeof(float));
  float*         A_buf  = (float*)(ws + (size_t)BB * HH * sizeof(float) + (size_t)BB * MM);

  // 1) occupation ranks (gather order)
  k_rank<<<dim3(BB / 256), dim3(256), 0, stream>>>(n, rank);

  // 2) h = tanh(n @ W1 + b1)   : grid (H/16/8, B/16)
  k_gemm1<<<dim3(HH / (16 * 8), BB / 16), dim3(256), 0, stream>>>(n, W1, b1, h_buf);

  // 3) A = gather(h @ W2 + b2) : grid (C/64/8, B/16), 16x64 macro-tiles
  k_gemm2<<<dim3(CC / (64 * 8), BB / 16), dim3(256), 0, stream>>>(h_buf, W2, b2, n, rank, A_buf);

  // 4) complex slogdet per 32x32
  k_slogdet<<<dim3(BB / 8), dim3(256), 0, stream>>>(A_buf, out);
}